// MaskedGraphAutoencoder_31190052503742
// MI455X (gfx1250) — compile-verified
//
#include <hip/hip_runtime.h>
#include <math.h>

typedef __attribute__((ext_vector_type(2))) float v2f;
typedef __attribute__((ext_vector_type(4))) float v4f;
typedef __attribute__((ext_vector_type(8))) float v8f;
typedef __attribute__((ext_vector_type(4))) int   v4i;

#define NEG_SLOPE 0.2f
#define GEMM_TB   256                 // threads per GEMM block (8 wave32)
#define GEMM_MW   8                   // M-tiles (16 rows each) per block
#define KMAX      512                 // max K across the three GEMMs

#if __has_builtin(__builtin_amdgcn_global_load_async_to_lds_b128)
#define HAVE_ASYNC_LDS 1
typedef __attribute__((address_space(1))) v4i* v4i_gptr;
typedef __attribute__((address_space(3))) v4i* v4i_lptr;
#else
#define HAVE_ASYNC_LDS 0
#endif

// ---------------------------------------------------------------------------
// float atomic max via int/uint monotonic bit trick (valid with -inf init)
// ---------------------------------------------------------------------------
__device__ inline void atomic_max_f32(float* addr, float val) {
  if (val >= 0.0f) {
    atomicMax((int*)addr, __float_as_int(val));
  } else {
    atomicMin((unsigned int*)addr, __float_as_uint(val));
  }
}

// ---------------------------------------------------------------------------
// C[M,Nc] = A[M,K] x B[K,Nc], f32 row-major, V_WMMA_F32_16X16X4_F32.
// Block = 8 waves: 8 consecutive 16-row M-tiles x one 16-col N-tile.
// The K x 16 slab of B is staged in LDS once per block (async global->LDS when
// the toolchain exposes the builtin), then each wave streams A from global
// (contiguous b64 per lane) and B fragments from LDS (conflict-free b32).
// Requires K % 4 == 0, Nc % 16 == 0 (true here: K in {256,512}, Nc in {512,256}).
// ---------------------------------------------------------------------------
__global__ void __launch_bounds__(GEMM_TB)
gemm_f32_wmma(const float* __restrict__ A, const float* __restrict__ B,
              float* __restrict__ C, int M, int K, int Nc) {
  __shared__ float Bs[KMAX * 16];

  const int ntN   = Nc >> 4;
  const int ntM   = (M + 15) >> 4;
  const int nblkM = (ntM + GEMM_MW - 1) / GEMM_MW;
  const int bm    = blockIdx.x % nblkM;        // M block
  const int nt    = blockIdx.x / nblkM;        // N tile
  const int tid   = threadIdx.x;
  (void)ntN;

  // ---- cooperative load of B[0:K, nt*16 : nt*16+16] into LDS ----
  // 4 threads per row (16B each), 64 rows per sweep.
  {
    const int r0 = tid >> 2;
    const int cc = (tid & 3) << 2;
    for (int r = r0; r < K; r += 64) {
      const float* gsrc = B + (size_t)r * Nc + (nt << 4) + cc;
      float*       ldst = &Bs[(r << 4) + cc];
#if HAVE_ASYNC_LDS
      __builtin_amdgcn_global_load_async_to_lds_b128(
          (v4i_gptr)(v4i*)gsrc, (v4i_lptr)(v4i*)ldst, 0, 0);
#else
      *(v4f*)ldst = *(const v4f*)gsrc;
#endif
    }
#if HAVE_ASYNC_LDS
#if __has_builtin(__builtin_amdgcn_s_wait_asynccnt)
    __builtin_amdgcn_s_wait_asynccnt(0);
#else
    asm volatile("s_wait_asynccnt 0x0" ::: "memory");
#endif
#endif
  }
  __syncthreads();

  // ---- per-wave 16x16 WMMA tile ----
  const int lane = tid & 31;
  const int mt   = bm * GEMM_MW + (tid >> 5);
  if (mt < ntM) {                               // wave-uniform: EXEC all-1 at WMMA
    int r = (mt << 4) + (lane & 15);
    if (r >= M) r = M - 1;                      // clamp tail loads (stores guarded)
    const int c16 = lane & 15;
    const int kg  = (lane >> 4) << 1;           // 0 or 2

    const float* __restrict__ Ap = A + (size_t)r * K;
    v8f acc = {};
    for (int k = 0; k < K; k += 4) {
      v2f a = *(const v2f*)(Ap + k + kg);       // contiguous pair -> b64 load
      v2f b;
      b.x = Bs[((k + kg)     << 4) + c16];
      b.y = Bs[((k + kg + 1) << 4) + c16];
      acc = __builtin_amdgcn_wmma_f32_16x16x4_f32(false, a, false, b,
                                                  (short)0, acc, false, false);
    }

    const int cg    = (nt << 4) + c16;
    const int rbase = (mt << 4) + ((lane >> 4) << 3);
#pragma unroll
    for (int v = 0; v < 8; ++v) {
      const int row = rbase + v;
      if (row < M) C[(size_t)row * Nc + cg] = acc[v];
    }
  }
}

// ---------------------------------------------------------------------------
// alpha_s / alpha_d per (node, head): wave-parallel dot product over C channels
// ---------------------------------------------------------------------------
__global__ void node_alpha(const float* __restrict__ h, const float* __restrict__ a_src,
                           const float* __restrict__ a_dst, float* __restrict__ as,
                           float* __restrict__ ad, int n, int heads, int C) {
  const int lane = threadIdx.x & 31;
  const int wave = blockIdx.x * (blockDim.x >> 5) + (threadIdx.x >> 5);
  if (wave >= n * heads) return;
  const int hd = wave % heads;
  const float* __restrict__ hp = h + (size_t)wave * C;   // wave == node*heads+hd
  float s = 0.0f, d = 0.0f;
  for (int ci = lane; ci < C; ci += 32) {
    const float v = hp[ci];
    s += v * a_src[hd * C + ci];
    d += v * a_dst[hd * C + ci];
  }
#pragma unroll
  for (int off = 16; off > 0; off >>= 1) {
    s += __shfl_down(s, off, 32);
    d += __shfl_down(d, off, 32);
  }
  if (lane == 0) { as[wave] = s; ad[wave] = d; }
}

__global__ void fill_f32(float* __restrict__ p, float v, size_t n) {
  const size_t i = (size_t)blockIdx.x * blockDim.x + threadIdx.x;
  if (i < n) p[i] = v;
}

// pass 1: e = leaky_relu(as[src]+ad[dst]); segment max over dst
__global__ void edge_logits_max(const long long* __restrict__ ei, int E, int n, int heads,
                                const float* __restrict__ as, const float* __restrict__ ad,
                                float* __restrict__ ew, float* __restrict__ mx) {
  const int i = blockIdx.x * blockDim.x + threadIdx.x;
  if (i >= E + n) return;
  const int s = (i < E) ? (int)ei[i]             : (i - E);
  const int d = (i < E) ? (int)ei[(size_t)E + i] : (i - E);
  for (int hd = 0; hd < heads; ++hd) {
    float e = as[s * heads + hd] + ad[d * heads + hd];
    e = (e >= 0.0f) ? e : NEG_SLOPE * e;
    ew[(size_t)i * heads + hd] = e;
    atomic_max_f32(&mx[d * heads + hd], e);
  }
}

// pass 2: w = exp(e - m[dst]); segment sum over dst
__global__ void edge_exp_sum(const long long* __restrict__ ei, int E, int n, int heads,
                             const float* __restrict__ mx, float* __restrict__ ew,
                             float* __restrict__ dn) {
  const int i = blockIdx.x * blockDim.x + threadIdx.x;
  if (i >= E + n) return;
  const int d = (i < E) ? (int)ei[(size_t)E + i] : (i - E);
  for (int hd = 0; hd < heads; ++hd) {
    const float w = expf(ew[(size_t)i * heads + hd] - mx[d * heads + hd]);
    ew[(size_t)i * heads + hd] = w;
    atomicAdd(&dn[d * heads + hd], w);
  }
}

// pass 3: out[dst] += h[src] * (w / (denom[dst]+eps)); one wave per edge
__global__ void edge_aggregate(const long long* __restrict__ ei, int E, int n, int heads,
                               int C, const float* __restrict__ h,
                               const float* __restrict__ ew, const float* __restrict__ dn,
                               float* __restrict__ out) {
  const int lane = threadIdx.x & 31;
  const int wave = blockIdx.x * (blockDim.x >> 5) + (threadIdx.x >> 5);
  if (wave >= E + n) return;
  const int s = (wave < E) ? (int)ei[wave]             : (wave - E);
  const int d = (wave < E) ? (int)ei[(size_t)E + wave] : (wave - E);
  const int F = heads * C;
  const float* __restrict__ hs = h + (size_t)s * F;
  float* __restrict__ od = out + (size_t)d * F;
  for (int f = lane; f < F; f += 32) {
    const int hd = f / C;
    const float att = ew[(size_t)wave * heads + hd] / (dn[d * heads + hd] + 1e-16f);
    atomicAdd(&od[f], hs[f] * att);
  }
}

// epilogue: += bias, optional PReLU
__global__ void bias_act(float* __restrict__ g, const float* __restrict__ b,
                         const float* __restrict__ pa, size_t total, int F, int do_prelu) {
  const size_t i = (size_t)blockIdx.x * blockDim.x + threadIdx.x;
  if (i >= total) return;
  float v = g[i] + b[i % F];
  if (do_prelu) { const float a = pa[0]; v = (v >= 0.0f) ? v : a * v; }
  g[i] = v;
}

// ---------------------------------------------------------------------------
static inline int cdiv_i(long long a, long long b) { return (int)((a + b - 1) / b); }

static inline int gemm_grid(int M, int Nc) {
  const int ntM = (M + 15) / 16;
  const int nblkM = (ntM + GEMM_MW - 1) / GEMM_MW;
  return nblkM * (Nc / 16);
}

extern "C" void kernel_launch(void* const* d_in, const int* in_sizes, int n_in,
                              void* d_out, int out_size, void* d_ws, size_t ws_size,
                              hipStream_t stream) {
  const float*     x       = (const float*)d_in[0];
  const long long* ei      = (const long long*)d_in[1];
  const float*     W1      = (const float*)d_in[2];
  const float*     a_src1  = (const float*)d_in[3];
  const float*     a_dst1  = (const float*)d_in[4];
  const float*     b1      = (const float*)d_in[5];
  const float*     prelu_a = (const float*)d_in[6];
  const float*     W2      = (const float*)d_in[7];
  const float*     a_src2  = (const float*)d_in[8];
  const float*     a_dst2  = (const float*)d_in[9];
  const float*     b2      = (const float*)d_in[10];
  const float*     Wd      = (const float*)d_in[11];
  const float*     a_srcd  = (const float*)d_in[12];
  const float*     a_dstd  = (const float*)d_in[13];
  const float*     bd      = (const float*)d_in[14];

  const int D = 256, F1 = 512, H1 = 4, C1 = 128, F2 = 256;
  const int n   = in_sizes[0] / D;   // 50000
  const int E   = in_sizes[1] / 2;   // 400000
  const int tot = E + n;             // edges incl. self loops

  float* ws = (float*)d_ws;
  size_t o = 0;
  float* h1 = ws + o; o += (size_t)n * F1;        // x@W1
  float* g1 = ws + o; o += (size_t)n * F1;        // layer-1 aggregate + PReLU
  float* h2 = ws + o; o += (size_t)n * F2;        // g1@W2
  float* as_ = ws + o; o += (size_t)n * H1;
  float* ad_ = ws + o; o += (size_t)n * H1;
  float* mx  = ws + o; o += (size_t)n * H1;
  float* dn  = ws + o; o += (size_t)n * H1;
  float* ew  = ws + o; o += (size_t)tot * H1;     // per-edge logits/weights
  float* h3 = h1;                                 // h1 dead after layer-1 aggregation

  float* z    = (float*)d_out;                    // (n, 256)
  float* xhat = (float*)d_out + (size_t)n * F2;   // (n, 256)

  const int   TB   = 256;                         // 8 wave32 per block
  const int   WPB  = TB / 32;
  const float NINF = -INFINITY;

  // ================= Layer 1: D -> (4 heads x 128), concat =================
  gemm_f32_wmma<<<gemm_grid(n, F1), GEMM_TB, 0, stream>>>(x, W1, h1, n, D, F1);
  node_alpha<<<cdiv_i((long long)n * H1, WPB), TB, 0, stream>>>(
      h1, a_src1, a_dst1, as_, ad_, n, H1, C1);
  fill_f32<<<cdiv_i((long long)n * H1, TB), TB, 0, stream>>>(mx, NINF, (size_t)n * H1);
  fill_f32<<<cdiv_i((long long)n * H1, TB), TB, 0, stream>>>(dn, 0.0f, (size_t)n * H1);
  fill_f32<<<cdiv_i((long long)n * F1, TB), TB, 0, stream>>>(g1, 0.0f, (size_t)n * F1);
  edge_logits_max<<<cdiv_i(tot, TB), TB, 0, stream>>>(ei, E, n, H1, as_, ad_, ew, mx);
  edge_exp_sum<<<cdiv_i(tot, TB), TB, 0, stream>>>(ei, E, n, H1, mx, ew, dn);
  edge_aggregate<<<cdiv_i(tot, WPB), TB, 0, stream>>>(ei, E, n, H1, C1, h1, ew, dn, g1);
  bias_act<<<cdiv_i((long long)n * F1, TB), TB, 0, stream>>>(
      g1, b1, prelu_a, (size_t)n * F1, F1, 1);

  // ================= Layer 2: 512 -> 256, 1 head, no concat ================
  gemm_f32_wmma<<<gemm_grid(n, F2), GEMM_TB, 0, stream>>>(g1, W2, h2, n, F1, F2);
  node_alpha<<<cdiv_i((long long)n, WPB), TB, 0, stream>>>(
      h2, a_src2, a_dst2, as_, ad_, n, 1, F2);
  fill_f32<<<cdiv_i(n, TB), TB, 0, stream>>>(mx, NINF, (size_t)n);
  fill_f32<<<cdiv_i(n, TB), TB, 0, stream>>>(dn, 0.0f, (size_t)n);
  fill_f32<<<cdiv_i((long long)n * F2, TB), TB, 0, stream>>>(z, 0.0f, (size_t)n * F2);
  edge_logits_max<<<cdiv_i(tot, TB), TB, 0, stream>>>(ei, E, n, 1, as_, ad_, ew, mx);
  edge_exp_sum<<<cdiv_i(tot, TB), TB, 0, stream>>>(ei, E, n, 1, mx, ew, dn);
  edge_aggregate<<<cdiv_i(tot, WPB), TB, 0, stream>>>(ei, E, n, 1, F2, h2, ew, dn, z);
  bias_act<<<cdiv_i((long long)n * F2, TB), TB, 0, stream>>>(
      z, b2, prelu_a, (size_t)n * F2, F2, 0);

  // ================= Decoder: 256 -> 256, 1 head, no concat ================
  gemm_f32_wmma<<<gemm_grid(n, F2), GEMM_TB, 0, stream>>>(z, Wd, h3, n, F2, F2);
  node_alpha<<<cdiv_i((long long)n, WPB), TB, 0, stream>>>(
      h3, a_srcd, a_dstd, as_, ad_, n, 1, F2);
  fill_f32<<<cdiv_i(n, TB), TB, 0, stream>>>(mx, NINF, (size_t)n);
  fill_f32<<<cdiv_i(n, TB), TB, 0, stream>>>(dn, 0.0f, (size_t)n);
  fill_f32<<<cdiv_i((long long)n * F2, TB), TB, 0, stream>>>(xhat, 0.0f, (size_t)n * F2);
  edge_logits_max<<<cdiv_i(tot, TB), TB, 0, stream>>>(ei, E, n, 1, as_, ad_, ew, mx);
  edge_exp_sum<<<cdiv_i(tot, TB), TB, 0, stream>>>(ei, E, n, 1, mx, ew, dn);
  edge_aggregate<<<cdiv_i(tot, WPB), TB, 0, stream>>>(ei, E, n, 1, F2, h3, ew, dn, xhat);
  bias_act<<<cdiv_i((long long)n * F2, TB), TB, 0, stream>>>(
      xhat, bd, prelu_a, (size_t)n * F2, F2, 0);
}